// LSTM_NATTEN_20761871909045
// MI455X (gfx1250) — compile-verified
//
#include <hip/hip_runtime.h>
#include <math.h>

// ---------------- problem constants (fixed by the reference) ----------------
#define TT   128          // time steps
#define PP   88           // parallel sequences / attention width
#define MROW (TT*PP)      // 11264 rows for all big GEMMs (multiple of 16)
#define IN0  133          // LSTM layer-0 input size
#define HH   24           // LSTM hidden
#define GG   96           // 4*H gates
#define CC   48           // channels (2*H)
#define NHEAD 4
#define HD   12           // head dim
#define KWIN 13
#define NNB  (KWIN*KWIN)  // 169
#define RPBW 25           // 2*KW-1

typedef __attribute__((ext_vector_type(2))) float v2f;
typedef __attribute__((ext_vector_type(8))) float v8f;

// ---------------------------------------------------------------------------
// Generic f32 WMMA GEMM: out[m,n] = sum_k A'[m,k] * W[n,k] + b1[n] + b2[n]
// MODE selects the row mapping of A (folds transposes into the GEMM):
//   MODE 0: row m            -> A + m*K                      (contiguous)
//   MODE 1: row m = n*T + t  -> A + (t*P + n)*K              (x: time-major -> seq-major)
//   MODE 2: row m = t*P + w  -> A + (w*T + t)*K              (h1: seq-major -> grid-major)
// KA: K is a multiple of 4 and rows are 8B-aligned -> unconditional b64 loads.
// One wave computes one 16x16 tile via V_WMMA_F32_16X16X4_F32 (K stepped by 4).
// M and N are multiples of 16 in every use.
// ---------------------------------------------------------------------------
template <int MODE, bool KA>
__global__ __launch_bounds__(256)
void gemm_wmma_f32(const float* __restrict__ A, const float* __restrict__ Wt,
                   const float* __restrict__ b1, const float* __restrict__ b2,
                   float* __restrict__ out, int M, int N, int K, int Tdim, int Pdim)
{
    const int lane = threadIdx.x & 31;
    const int wid  = threadIdx.x >> 5;
    const int tile = blockIdx.x * (blockDim.x >> 5) + wid;
    const int Mtiles = M >> 4;
    const int Ntiles = N >> 4;
    if (tile >= Mtiles * Ntiles) return;              // wave-uniform guard
    const int mt = tile % Mtiles;
    const int nt = tile / Mtiles;
    const int mBase = mt << 4;
    const int nBase = nt << 4;

    // A row for this lane (lanes 0-15 and 16-31 both cover rows M=0..15)
    const int m = mBase + (lane & 15);
    const float* arow;
    if (MODE == 0) {
        arow = A + (size_t)m * K;
    } else if (MODE == 1) {
        const int n_ = m / Tdim, t_ = m - n_ * Tdim;
        arow = A + (size_t)(t_ * Pdim + n_) * K;
    } else {
        const int t_ = m / Pdim, w_ = m - t_ * Pdim;
        arow = A + (size_t)(w_ * Tdim + t_) * K;
    }
    const float* wrow = Wt + (size_t)(nBase + (lane & 15)) * K;

    v8f acc = {0.f, 0.f, 0.f, 0.f, 0.f, 0.f, 0.f, 0.f};
    const int half = lane >> 4;                       // upper half-wave holds K+2,K+3

    if (KA) {
        // K % 4 == 0, rows 8B aligned: straight global_load_b64 per fragment.
        const v2f* a2 = (const v2f*)arow;
        const v2f* b2 = (const v2f*)wrow;
        for (int kk = 0; kk < K; kk += 4) {
            __builtin_prefetch(arow + kk + 32, 0, 1); // global_prefetch_b8
            const v2f a = a2[(kk >> 1) + half];
            const v2f b = b2[(kk >> 1) + half];
            acc = __builtin_amdgcn_wmma_f32_16x16x4_f32(
                      false, a, false, b, (short)0, acc, false, false);
        }
    } else {
        // Ragged K: clamp addresses (always-valid unconditional loads) and
        // zero the tail with VALU cndmask -- no EXEC-masked load branches.
        for (int kk = 0; kk < K; kk += 4) {
            const int k0 = kk + (half << 1);
            const int kc0 = (k0     < K) ? k0     : 0;
            const int kc1 = (k0 + 1 < K) ? k0 + 1 : 0;
            __builtin_prefetch(arow + kk + 32, 0, 1);
            v2f a, b;
            a.x = arow[kc0]; a.y = arow[kc1];
            b.x = wrow[kc0]; b.y = wrow[kc1];
            a.x = (k0     < K) ? a.x : 0.0f;
            a.y = (k0 + 1 < K) ? a.y : 0.0f;
            b.x = (k0     < K) ? b.x : 0.0f;
            b.y = (k0 + 1 < K) ? b.y : 0.0f;
            acc = __builtin_amdgcn_wmma_f32_16x16x4_f32(
                      false, a, false, b, (short)0, acc, false, false);
        }
    }

    const int col   = nBase + (lane & 15);
    const float bia = (b1 ? b1[col] : 0.0f) + (b2 ? b2[col] : 0.0f);
    const int rbase = mBase + (half << 3);            // lanes>=16 hold rows M+8..M+15
    #pragma unroll
    for (int r = 0; r < 8; ++r)
        out[(size_t)(rbase + r) * N + col] = acc[r] + bia;
}

// ---------------------------------------------------------------------------
// LSTM recurrence: one block per (sequence, direction), 96 threads (3 waves).
// Thread g owns gate g; per-gate w_hh row (24 floats) in registers; h/c in LDS.
// Backward direction walks t = 127..0 so no explicit flips are needed.
// ---------------------------------------------------------------------------
__device__ __forceinline__ float sigm(float x) { return 1.0f / (1.0f + __expf(-x)); }

__global__ __launch_bounds__(GG)
void lstm_rec(const float* __restrict__ xgF, const float* __restrict__ xgB,
              const float* __restrict__ whF, const float* __restrict__ whB,
              float* __restrict__ hout)
{
    const int n   = blockIdx.x;        // sequence 0..87
    const int dir = blockIdx.y;        // 0 fwd, 1 bwd
    const int g   = threadIdx.x;       // gate 0..95
    const float* xg = dir ? xgB : xgF;
    const float* wh = dir ? whB : whF;

    __shared__ float h_s[HH], c_s[HH], gbuf[GG];
    float wrow[HH];
    #pragma unroll
    for (int j = 0; j < HH; ++j) wrow[j] = wh[g * HH + j];
    if (g < HH) { h_s[g] = 0.f; c_s[g] = 0.f; }
    __syncthreads();

    for (int s = 0; s < TT; ++s) {
        const int t = dir ? (TT - 1 - s) : s;
        float pre = xg[(n * TT + t) * GG + g];
        #pragma unroll
        for (int j = 0; j < HH; ++j) pre += h_s[j] * wrow[j];
        gbuf[g] = pre;
        __syncthreads();
        if (g < HH) {
            const float ig = sigm(gbuf[g]);
            const float fg = sigm(gbuf[HH + g]);
            const float gt = tanhf(gbuf[2 * HH + g]);
            const float og = sigm(gbuf[3 * HH + g]);
            const float c  = fg * c_s[g] + ig * gt;
            c_s[g] = c;
            const float h = og * tanhf(c);
            h_s[g] = h;
            hout[(n * TT + t) * CC + dir * HH + g] = h;
        }
        __syncthreads();
    }
}

// ---------------------------------------------------------------------------
// 13x13 neighborhood attention: one block per grid position, one wave per head.
// Lanes split the 169 neighbors; wave32 shfl_xor butterflies for softmax + V.
// qkv row layout: [q(48) | k(48) | v(48)], rows ordered m = t*88 + w.
// ---------------------------------------------------------------------------
__global__ __launch_bounds__(NHEAD * 32)
void natten2d(const float* __restrict__ qkv, const float* __restrict__ rpb,
              float* __restrict__ out)
{
    const int pos  = blockIdx.x;             // t*PP + w
    const int t    = pos / PP;
    const int w    = pos - t * PP;
    const int head = threadIdx.x >> 5;
    const int lane = threadIdx.x & 31;

    const int it0 = min(max(t - (KWIN / 2), 0), TT - KWIN);
    const int iw0 = min(max(w - (KWIN / 2), 0), PP - KWIN);

    const float* qp = qkv + (size_t)pos * (3 * CC) + head * HD;
    float q[HD];
    #pragma unroll
    for (int d = 0; d < HD; ++d) q[d] = qp[d] * 0.28867513459481287f; // 12^-0.5

    // logits for this lane's neighbors (<=6 of the 169)
    float logit[6];
    float mx = -1e30f;
    #pragma unroll
    for (int j = 0; j < 6; ++j) {
        const int idx = lane + j * 32;
        float l = -1e30f;
        if (idx < NNB) {
            const int p_ = idx / KWIN, q_ = idx - p_ * KWIN;
            const int kt = it0 + p_,   kw = iw0 + q_;
            const float* kp = qkv + (size_t)(kt * PP + kw) * (3 * CC) + CC + head * HD;
            float acc = rpb[head * (RPBW * RPBW) + (kt - t + KWIN - 1) * RPBW + (kw - w + KWIN - 1)];
            #pragma unroll
            for (int d = 0; d < HD; ++d) acc += q[d] * kp[d];
            l = acc;
        }
        logit[j] = l;
        mx = fmaxf(mx, l);
    }
    #pragma unroll
    for (int off = 16; off > 0; off >>= 1) mx = fmaxf(mx, __shfl_xor(mx, off, 32));

    float vout[HD];
    #pragma unroll
    for (int d = 0; d < HD; ++d) vout[d] = 0.f;
    float ssum = 0.f;
    #pragma unroll
    for (int j = 0; j < 6; ++j) {
        const int idx = lane + j * 32;
        if (idx < NNB) {
            const float e = __expf(logit[j] - mx);
            ssum += e;
            const int p_ = idx / KWIN, q_ = idx - p_ * KWIN;
            const float* vp = qkv + (size_t)((it0 + p_) * PP + (iw0 + q_)) * (3 * CC) + 2 * CC + head * HD;
            #pragma unroll
            for (int d = 0; d < HD; ++d) vout[d] += e * vp[d];
        }
    }
    #pragma unroll
    for (int off = 16; off > 0; off >>= 1) {
        ssum += __shfl_xor(ssum, off, 32);
        #pragma unroll
        for (int d = 0; d < HD; ++d) vout[d] += __shfl_xor(vout[d], off, 32);
    }
    if (lane == 0) {
        const float inv = 1.0f / ssum;
        float* op = out + (size_t)pos * CC + head * HD;
        #pragma unroll
        for (int d = 0; d < HD; ++d) op[d] = vout[d] * inv;
    }
}

// ---------------------------------------------------------------------------
extern "C" void kernel_launch(void* const* d_in, const int* in_sizes, int n_in,
                              void* d_out, int out_size, void* d_ws, size_t ws_size,
                              hipStream_t stream)
{
    const float* x        = (const float*)d_in[0];
    const float* w_ih_l0  = (const float*)d_in[1];
    const float* w_hh_l0  = (const float*)d_in[2];
    const float* b_ih_l0  = (const float*)d_in[3];
    const float* b_hh_l0  = (const float*)d_in[4];
    const float* w_ih_l0r = (const float*)d_in[5];
    const float* w_hh_l0r = (const float*)d_in[6];
    const float* b_ih_l0r = (const float*)d_in[7];
    const float* b_hh_l0r = (const float*)d_in[8];
    const float* w_ih_l1  = (const float*)d_in[9];
    const float* w_hh_l1  = (const float*)d_in[10];
    const float* b_ih_l1  = (const float*)d_in[11];
    const float* b_hh_l1  = (const float*)d_in[12];
    const float* w_ih_l1r = (const float*)d_in[13];
    const float* w_hh_l1r = (const float*)d_in[14];
    const float* b_ih_l1r = (const float*)d_in[15];
    const float* b_hh_l1r = (const float*)d_in[16];
    const float* w_qkv    = (const float*)d_in[17];
    const float* b_qkv    = (const float*)d_in[18];
    const float* rpb      = (const float*)d_in[19];
    const float* w_proj   = (const float*)d_in[20];
    const float* b_proj   = (const float*)d_in[21];
    float* outp = (float*)d_out;

    // workspace layout (floats)
    const size_t XG_SZ = (size_t)MROW * GG;        // 1,081,344
    const size_t H_SZ  = (size_t)MROW * CC;        //   540,672
    const size_t Q_SZ  = (size_t)MROW * 3 * CC;    // 1,622,016
    float* ws   = (float*)d_ws;
    float* xgA  = ws;
    float* xgB  = xgA + XG_SZ;
    float* h0   = xgB + XG_SZ;
    float* h1   = h0  + H_SZ;
    float* qkvb = h1  + H_SZ;
    float* att  = qkvb + Q_SZ;
    float* yb   = att  + H_SZ;
    const size_t need = (2 * XG_SZ + 2 * H_SZ + Q_SZ + 2 * H_SZ) * sizeof(float);
    if (ws_size < need) return;

    const int waves_per_block = 8;                 // 256 threads
    auto tiles_grid = [&](int N) {
        const int tiles = (MROW >> 4) * (N >> 4);
        return dim3((tiles + waves_per_block - 1) / waves_per_block);
    };

    // ---- BiLSTM layer 0: batched input projections (K=133, ragged) ----
    gemm_wmma_f32<1, false><<<tiles_grid(GG), 256, 0, stream>>>(x, w_ih_l0,  b_ih_l0,  b_hh_l0,  xgA, MROW, GG, IN0, TT, PP);
    gemm_wmma_f32<1, false><<<tiles_grid(GG), 256, 0, stream>>>(x, w_ih_l0r, b_ih_l0r, b_hh_l0r, xgB, MROW, GG, IN0, TT, PP);
    lstm_rec<<<dim3(PP, 2), GG, 0, stream>>>(xgA, xgB, w_hh_l0, w_hh_l0r, h0);

    // ---- BiLSTM layer 1 (K=48, aligned) ----
    gemm_wmma_f32<0, true><<<tiles_grid(GG), 256, 0, stream>>>(h0, w_ih_l1,  b_ih_l1,  b_hh_l1,  xgA, MROW, GG, CC, 0, 0);
    gemm_wmma_f32<0, true><<<tiles_grid(GG), 256, 0, stream>>>(h0, w_ih_l1r, b_ih_l1r, b_hh_l1r, xgB, MROW, GG, CC, 0, 0);
    lstm_rec<<<dim3(PP, 2), GG, 0, stream>>>(xgA, xgB, w_hh_l1, w_hh_l1r, h1);

    // ---- NATTEN pass 1 (transpose h1 -> grid order folded into qkv GEMM) ----
    gemm_wmma_f32<2, true><<<tiles_grid(3 * CC), 256, 0, stream>>>(h1, w_qkv, b_qkv, nullptr, qkvb, MROW, 3 * CC, CC, TT, PP);
    natten2d<<<MROW, NHEAD * 32, 0, stream>>>(qkvb, rpb, att);
    gemm_wmma_f32<0, true><<<tiles_grid(CC), 256, 0, stream>>>(att, w_proj, b_proj, nullptr, yb, MROW, CC, CC, 0, 0);

    // ---- NATTEN pass 2 (writes final output) ----
    gemm_wmma_f32<0, true><<<tiles_grid(3 * CC), 256, 0, stream>>>(yb, w_qkv, b_qkv, nullptr, qkvb, MROW, 3 * CC, CC, 0, 0);
    natten2d<<<MROW, NHEAD * 32, 0, stream>>>(qkvb, rpb, att);
    gemm_wmma_f32<0, true><<<tiles_grid(CC), 256, 0, stream>>>(att, w_proj, b_proj, nullptr, outp, MROW, CC, CC, 0, 0);
}